// XLSTMSocialPoolingLayer_19207093748279
// MI455X (gfx1250) — compile-verified
//
#include <hip/hip_runtime.h>

typedef float v8f __attribute__((ext_vector_type(8)));
typedef float v2f __attribute__((ext_vector_type(2)));

#define N_AGENTS 256
#define BATCH 64
#define SEQ 20
#define HID 128
#define MASK_STRIDE 260   // 256 + 4 dword pad -> conflict-free ds_load_b64 of A-tiles

__global__ __launch_bounds__(256)
void social_pool_wmma_f32(const float* __restrict__ hs,   // (256,64,20,128)
                          const float* __restrict__ pos,  // (256,64,20,2)
                          const float* __restrict__ radius,
                          float* __restrict__ out)        // (256,64,128)
{
    __shared__ __align__(16) float s_mask[16 * MASK_STRIDE]; // 16.6 KB
    __shared__ __align__(16) float s_pos[N_AGENTS * 2];      // 2 KB
    __shared__ float s_cnt[16];

    const int tid = threadIdx.x;
    const int it  = blockIdx.x;          // i-tile (16 agents per tile)
    const int b   = blockIdx.y;          // batch
    const int i0  = it * 16;

    // ---- stage last-timestep positions of all 256 agents for this batch ----
    {
        const int j = tid;               // 0..255
        const float* p = pos + ((size_t)(j * BATCH + b) * SEQ + (SEQ - 1)) * 2;
        s_pos[2 * j + 0] = p[0];
        s_pos[2 * j + 1] = p[1];
        if (tid < 16) s_cnt[tid] = 0.0f;
    }
    __syncthreads();

    const float r = radius[0];

    // ---- build 16x256 mask tile in LDS + per-row counts ----
    {
        const int il = tid & 15;         // local i row 0..15
        const int jb = tid >> 4;         // 0..15
        const float pix = s_pos[2 * (i0 + il) + 0];
        const float piy = s_pos[2 * (i0 + il) + 1];
        float cacc = 0.0f;
        #pragma unroll
        for (int s = 0; s < 16; ++s) {
            const int j = jb + 16 * s;
            const float dx = pix - s_pos[2 * j + 0];
            const float dy = piy - s_pos[2 * j + 1];
            const float d  = sqrtf(dx * dx + dy * dy);   // match reference sqrt-then-compare
            const float m  = ((d <= r) && ((i0 + il) != j)) ? 1.0f : 0.0f;
            s_mask[il * MASK_STRIDE + j] = m;
            cacc += m;
        }
        atomicAdd(&s_cnt[il], cacc);     // ds_add_f32
    }
    __syncthreads();

    // ---- 16(i) x 16(h) output tile per wave via V_WMMA_F32_16X16X4_F32 ----
    const int lane = tid & 31;
    const int wave = tid >> 5;           // 0..7 -> HID chunk
    const int hc   = wave * 16;
    const int n    = lane & 15;          // N column within tile
    const int half = lane >> 4;          // 0: K=0,1  1: K=2,3 (per ISA 16x4 f32 layout)

    // B source: h_last[j][b][hc+n]; j-stride in the (N,B,S,H) tensor
    const float* hb = hs + ((size_t)b * SEQ + (SEQ - 1)) * HID + hc + n;
    const size_t jstride = (size_t)BATCH * SEQ * HID;    // 163840 floats

    // A source: mask[m][j0 + 2*half .. +1], m = lane & 15
    const float* amask = &s_mask[(lane & 15) * MASK_STRIDE + (half << 1)];

    v8f acc = {};
    #pragma unroll 4
    for (int j0 = 0; j0 < N_AGENTS; j0 += 4) {
        v2f a = *(const v2f*)(amask + j0);               // ds_load_b64, conflict-free
        const int krow = j0 + (half << 1);
        v2f bv;
        bv.x = hb[(size_t)krow * jstride];
        bv.y = hb[(size_t)(krow + 1) * jstride];
        // 8 args: (neg_a, A, neg_b, B, c_mod, C, reuse_a, reuse_b)
        acc = __builtin_amdgcn_wmma_f32_16x16x4_f32(
            false, a, false, bv, (short)0, acc, false, false);
    }

    // ---- scale rows by 1/max(count,1) and store (D layout: M = v + 8*half, N = lane&15)
    #pragma unroll
    for (int v = 0; v < 8; ++v) {
        const int M = v + (half << 3);
        float c = s_cnt[M];
        c = (c > 1.0f) ? c : 1.0f;
        out[(((size_t)(i0 + M) * BATCH) + b) * HID + hc + n] = acc[v] / c;
    }
}

extern "C" void kernel_launch(void* const* d_in, const int* in_sizes, int n_in,
                              void* d_out, int out_size, void* d_ws, size_t ws_size,
                              hipStream_t stream) {
    (void)in_sizes; (void)n_in; (void)out_size; (void)d_ws; (void)ws_size;
    const float* hs  = (const float*)d_in[0];
    const float* pos = (const float*)d_in[1];
    const float* rad = (const float*)d_in[2];
    float* out = (float*)d_out;

    dim3 grid(16, 64);   // (i-tiles, batches)
    dim3 block(256);     // 8 wave32 -> 8 HID chunks of 16
    hipLaunchKernelGGL(social_pool_wmma_f32, grid, block, 0, stream, hs, pos, rad, out);
}